// NORMMSSSIM_11072425689377
// MI455X (gfx1250) — compile-verified
//
#include <hip/hip_runtime.h>
#include <math.h>

// ---------------------------------------------------------------------------
// MS-SSIM (5 dilated levels) for 16x1x1024x1024 fp32 images on gfx1250.
// Column convolution runs on the matrix pipe: V_WMMA_F32_16X16X4_F32 with a
// banded 16x(16+4d) weight matrix against the row-convolved halo buffer.
// Kernels are templated on dilation so band weights / loop trip counts /
// LDS footprints are compile-time constants. All index math is 32-bit
// shift/mask (every level is a square power of two).
// ---------------------------------------------------------------------------

typedef __attribute__((ext_vector_type(2))) float v2f;
typedef __attribute__((ext_vector_type(8))) float v8f;

#define C1F 6.5025f    // (0.01*255)^2
#define C2F 58.5225f   // (0.03*255)^2
#define EPSF 1e-12f

// Normalized 1-D Gaussian, window=5, sigma=1 (fp64-derived, folds to
// inline constants).
#define G5_INIT {0.05448868454964294f, 0.24420134203116870f, \
                 0.40261994683837680f, 0.24420134203116870f, \
                 0.05448868454964294f}

// Banded vertical-conv weight W[m, rr]: out[m] = sum_k g[k]*halo[m + D*k].
// With D compile-time this is 5 compare-selects, no division.
template <int D>
__device__ __forceinline__ float bandW(int m, int rr) {
  constexpr float g[5] = G5_INIT;
  const int c = rr - m;
  float w = 0.0f;
#pragma unroll
  for (int k = 0; k < 5; ++k) w = (c == k * D) ? g[k] : w;
  return w;
}

// ---------------------------------------------------------------------------
// Global max of img2 (two-stage, deterministic, b128 loads)
// ---------------------------------------------------------------------------
__global__ void max_stage1(const float4* __restrict__ x, int n4,
                           float* __restrict__ partial) {
  __shared__ float sm[256];
  float v = -1e30f;
  for (int i = blockIdx.x * 256 + threadIdx.x; i < n4; i += gridDim.x * 256) {
    const float4 t = x[i];
    v = fmaxf(v, fmaxf(fmaxf(t.x, t.y), fmaxf(t.z, t.w)));
  }
  sm[threadIdx.x] = v;
  __syncthreads();
  for (int s = 128; s; s >>= 1) {
    if (threadIdx.x < (unsigned)s)
      sm[threadIdx.x] = fmaxf(sm[threadIdx.x], sm[threadIdx.x + s]);
    __syncthreads();
  }
  if (threadIdx.x == 0) partial[blockIdx.x] = sm[0];
}

__global__ void max_stage2(const float* __restrict__ partial, int n,
                           float* __restrict__ nrm) {
  __shared__ float sm[256];
  float v = -1e30f;
  for (int i = threadIdx.x; i < n; i += 256) v = fmaxf(v, partial[i]);
  sm[threadIdx.x] = v;
  __syncthreads();
  for (int s = 128; s; s >>= 1) {
    if (threadIdx.x < (unsigned)s)
      sm[threadIdx.x] = fmaxf(sm[threadIdx.x], sm[threadIdx.x + s]);
    __syncthreads();
  }
  if (threadIdx.x == 0) nrm[0] = 255.0f / (sm[0] + EPSF);
}

// ---------------------------------------------------------------------------
// SSIM level kernel: one wave32 per 16x16 output tile, templated on dilation.
//  - LDS halo patch (zero outside image == conv zero padding)
//  - VALU separable row conv of {a, b, a*a, b*b, a*b}
//  - WMMA f32 16x16x4 column conv (banded A per ISA A-layout; T from LDS)
//  - SSIM map + wave shuffle reduce -> one partial per tile
// ---------------------------------------------------------------------------
template <int DIL>
__global__ __launch_bounds__(32) void ssim_level(
    const float* __restrict__ A, const float* __restrict__ B, int H, int W,
    const float* __restrict__ nrm, int doNorm, float* __restrict__ partial) {
  constexpr int P = 16 + 4 * DIL;   // halo patch dimension
  constexpr int HALO = 2 * DIL;
  constexpr int NCH = P / 4;        // K-chunks of 4 for WMMA

  __shared__ float sA[P * P];
  __shared__ float sB[P * P];
  __shared__ float sRow[5][P * 16];  // 5 row-convolved quantities

  constexpr float g[5] = G5_INIT;

  const int lane = threadIdx.x;
  const int base = (int)blockIdx.z * H * W;  // <= 16M, fits int
  const int r0 = (int)blockIdx.y * 16 - HALO;
  const int c0 = (int)blockIdx.x * 16 - HALO;
  const float s255 = nrm[0];

  // --- load zero-padded (and optionally normalized) patch into LDS ---
  for (int idx = lane; idx < P * P; idx += 32) {
    const int r = idx / P, c = idx - r * P;  // constant divisor
    const int gr = r0 + r, gc = c0 + c;
    float va = 0.0f, vb = 0.0f;
    if (gr >= 0 && gr < H && gc >= 0 && gc < W) {
      const int o = base + gr * W + gc;
      va = A[o];
      vb = B[o];
      if (doNorm) {
        va = (va + EPSF) * s255;
        vb = (vb + EPSF) * s255;
      }
    }
    sA[idx] = va;
    sB[idx] = vb;
  }
  __syncthreads();

  // --- horizontal (row) conv of the 5 quantities over all halo rows ---
#pragma unroll
  for (int idx = lane; idx < P * 16; idx += 32) {
    const int r = idx >> 4, c = idx & 15;
    float m1 = 0.f, m2 = 0.f, p11 = 0.f, p22 = 0.f, p12 = 0.f;
#pragma unroll
    for (int k = 0; k < 5; ++k) {
      const float w = g[k];
      const float a = sA[r * P + c + DIL * k];
      const float b = sB[r * P + c + DIL * k];
      m1 += w * a;
      m2 += w * b;
      p11 += w * a * a;
      p22 += w * b * b;
      p12 += w * a * b;
    }
    sRow[0][idx] = m1;
    sRow[1][idx] = m2;
    sRow[2][idx] = p11;
    sRow[3][idx] = p22;
    sRow[4][idx] = p12;
  }
  __syncthreads();

  // --- vertical conv via WMMA: out = sum_j Aband_j(16x4) x T_j(4x16) ---
  v8f acc[5];
#pragma unroll
  for (int q = 0; q < 5; ++q)
#pragma unroll
    for (int i = 0; i < 8; ++i) acc[q][i] = 0.0f;

  const int m = lane & 15;
  const int koff = (lane >> 4) << 1;  // lanes 0-15: K=0/1, lanes 16-31: K=2/3
#pragma unroll
  for (int j = 0; j < NCH; ++j) {
    const int rr0 = 4 * j + koff;
    v2f af;
    af.x = bandW<DIL>(m, rr0);
    af.y = bandW<DIL>(m, rr0 + 1);
#pragma unroll
    for (int q = 0; q < 5; ++q) {
      v2f bf;
      bf.x = sRow[q][rr0 * 16 + m];
      bf.y = sRow[q][(rr0 + 1) * 16 + m];
      acc[q] = __builtin_amdgcn_wmma_f32_16x16x4_f32(
          false, af, false, bf, (short)0, acc[q], false, false);
    }
  }

  // --- SSIM map on the 8 per-lane C/D elements, then wave reduce ---
  float lsum = 0.0f;
#pragma unroll
  for (int i = 0; i < 8; ++i) {
    const float mu1 = acc[0][i], mu2 = acc[1][i];
    const float mu1s = mu1 * mu1, mu2s = mu2 * mu2, mu12 = mu1 * mu2;
    const float sg1 = acc[2][i] - mu1s;
    const float sg2 = acc[3][i] - mu2s;
    const float sg12 = acc[4][i] - mu12;
    const float num = (2.0f * mu12 + C1F) * (2.0f * sg12 + C2F);
    const float den = (mu1s + mu2s + C1F) * (sg1 + sg2 + C2F);
    lsum += num / den;
  }
  for (int off = 16; off; off >>= 1) lsum += __shfl_xor(lsum, off, 32);

  if (lane == 0) {
    const int tile =
        ((int)blockIdx.z * gridDim.y + blockIdx.y) * gridDim.x + blockIdx.x;
    partial[tile] = lsum;
  }
}

// ---------------------------------------------------------------------------
// 2x2 average pool (optionally fused normalization at level 0).
// Square power-of-two levels: index math is mask/shift; b64 row loads.
// ---------------------------------------------------------------------------
__global__ void pool2(const float* __restrict__ A, const float* __restrict__ B,
                      float* __restrict__ Ao, float* __restrict__ Bo, int Hi,
                      int Wi, int logWo, const float* __restrict__ nrm,
                      int doNorm) {
  const int Wo = Wi >> 1;
  const int n = 16 * (Hi >> 1) * Wo;        // <= 4.2M, fits int
  const int rowPitch2 = Wi >> 1;            // one input row, in float2 units
  const int imgPitch2 = (Hi * Wi) >> 1;     // one input image, in float2 units
  const float s255 = nrm[0];
  const float2* __restrict__ A2 = (const float2*)A;
  const float2* __restrict__ B2 = (const float2*)B;
  for (int i = blockIdx.x * blockDim.x + threadIdx.x; i < n;
       i += gridDim.x * blockDim.x) {
    const int x = i & (Wo - 1);
    const int y = (i >> logWo) & (Wo - 1);  // Ho == Wo (square levels)
    const int img = i >> (2 * logWo);
    const int t0 = img * imgPitch2 + y * Wi + x;  // float2 idx of row 2y
    const int t1 = t0 + rowPitch2;                // row 2y+1
    float2 at = A2[t0], ab = A2[t1];
    float2 bt = B2[t0], bb = B2[t1];
    if (doNorm) {
      at.x = (at.x + EPSF) * s255; at.y = (at.y + EPSF) * s255;
      ab.x = (ab.x + EPSF) * s255; ab.y = (ab.y + EPSF) * s255;
      bt.x = (bt.x + EPSF) * s255; bt.y = (bt.y + EPSF) * s255;
      bb.x = (bb.x + EPSF) * s255; bb.y = (bb.y + EPSF) * s255;
    }
    Ao[i] = 0.25f * ((at.x + at.y) + (ab.x + ab.y));
    Bo[i] = 0.25f * ((bt.x + bt.y) + (bb.x + bb.y));
  }
}

// ---------------------------------------------------------------------------
// Finalize: deterministic sums of per-tile partials -> means -> scalar
// ---------------------------------------------------------------------------
__global__ void finalize_k(const float* __restrict__ ws,
                           const float* __restrict__ wts,
                           float* __restrict__ out) {
  __shared__ float sm[256];
  __shared__ float means[5];
  const int bases[5] = {2048, 67584, 83968, 88064, 89088};
  const int cnts[5] = {65536, 16384, 4096, 1024, 256};
  const float numel[5] = {16777216.0f, 4194304.0f, 1048576.0f, 262144.0f,
                          65536.0f};
  for (int s = 0; s < 5; ++s) {
    float v = 0.0f;
    for (int i = threadIdx.x; i < cnts[s]; i += 256) v += ws[bases[s] + i];
    sm[threadIdx.x] = v;
    __syncthreads();
    for (int st = 128; st; st >>= 1) {
      if (threadIdx.x < (unsigned)st) sm[threadIdx.x] += sm[threadIdx.x + st];
      __syncthreads();
    }
    if (threadIdx.x == 0) means[s] = sm[0] / numel[s];
    __syncthreads();
  }
  if (threadIdx.x == 0) {
    float p = 1.0f;
    for (int s = 0; s < 5; ++s) p *= powf(means[s], wts[s]);
    out[0] = 1.0f - p;
  }
}

// ---------------------------------------------------------------------------
// Workspace layout (float offsets)
// ---------------------------------------------------------------------------
#define WS_MAXP 0        // 1024
#define WS_NRM 1024      // 1
#define WS_P0 2048       // 65536
#define WS_P1 67584      // 16384
#define WS_P2 83968      // 4096
#define WS_P3 88064      // 1024
#define WS_P4 89088      // 256
#define WS_A1 131072     // 4194304
#define WS_B1 4325376
#define WS_A2 8519680    // 1048576
#define WS_B2 9568256
#define WS_A3 10616832   // 262144
#define WS_B3 10878976
#define WS_A4 11141120   // 65536
#define WS_B4 11206656

extern "C" void kernel_launch(void* const* d_in, const int* in_sizes, int n_in,
                              void* d_out, int out_size, void* d_ws,
                              size_t ws_size, hipStream_t stream) {
  const float* img1 = (const float*)d_in[0];
  const float* img2 = (const float*)d_in[1];
  // d_in[2] = 5x5 window (separable Gaussian, folded to constants above)
  const float* weights = (const float*)d_in[3];
  float* W = (float*)d_ws;
  float* out = (float*)d_out;

  // normalization scale (16M elements as 4M float4)
  max_stage1<<<1024, 256, 0, stream>>>((const float4*)img2, 4194304,
                                       W + WS_MAXP);
  max_stage2<<<1, 256, 0, stream>>>(W + WS_MAXP, 1024, W + WS_NRM);

  // level 0 (fused normalize), dilation 1
  ssim_level<1><<<dim3(64, 64, 16), 32, 0, stream>>>(img1, img2, 1024, 1024,
                                                     W + WS_NRM, 1, W + WS_P0);
  pool2<<<2048, 256, 0, stream>>>(img1, img2, W + WS_A1, W + WS_B1, 1024, 1024,
                                  9, W + WS_NRM, 1);
  // level 1, dilation 2
  ssim_level<2><<<dim3(32, 32, 16), 32, 0, stream>>>(
      W + WS_A1, W + WS_B1, 512, 512, W + WS_NRM, 0, W + WS_P1);
  pool2<<<1024, 256, 0, stream>>>(W + WS_A1, W + WS_B1, W + WS_A2, W + WS_B2,
                                  512, 512, 8, W + WS_NRM, 0);
  // level 2, dilation 3
  ssim_level<3><<<dim3(16, 16, 16), 32, 0, stream>>>(
      W + WS_A2, W + WS_B2, 256, 256, W + WS_NRM, 0, W + WS_P2);
  pool2<<<256, 256, 0, stream>>>(W + WS_A2, W + WS_B2, W + WS_A3, W + WS_B3,
                                 256, 256, 7, W + WS_NRM, 0);
  // level 3, dilation 6
  ssim_level<6><<<dim3(8, 8, 16), 32, 0, stream>>>(
      W + WS_A3, W + WS_B3, 128, 128, W + WS_NRM, 0, W + WS_P3);
  pool2<<<64, 256, 0, stream>>>(W + WS_A3, W + WS_B3, W + WS_A4, W + WS_B4, 128,
                                128, 6, W + WS_NRM, 0);
  // level 4, dilation 9
  ssim_level<9><<<dim3(4, 4, 16), 32, 0, stream>>>(
      W + WS_A4, W + WS_B4, 64, 64, W + WS_NRM, 0, W + WS_P4);

  finalize_k<<<1, 256, 0, stream>>>(W, weights, out);
}